// ChannelAttentionLayer_2379411882261
// MI455X (gfx1250) — compile-verified
//
#include <hip/hip_runtime.h>
#include <math.h>

// ---- problem constants -----------------------------------------------------
#define B_   32
#define H_   56
#define W_   56
#define C_   256
#define HW_  (H_ * W_)       // 3136
#define CR_  32              // C / reduction(8)
#define S_   16              // HW splits per batch for pooling
#define ROWS_ (HW_ / S_)     // 196 rows per partial block

typedef float f4  __attribute__((ext_vector_type(4)));
typedef float v2f __attribute__((ext_vector_type(2)));
typedef float v8f __attribute__((ext_vector_type(8)));

__device__ __forceinline__ f4 max4(f4 a, f4 b) {
    f4 r;
    r.x = fmaxf(a.x, b.x); r.y = fmaxf(a.y, b.y);
    r.z = fmaxf(a.z, b.z); r.w = fmaxf(a.w, b.w);
    return r;
}

// ---- kernel 1: partial avg/max pool ---------------------------------------
// grid (B, S), block 256. Thread t: channel-group c4 = t&63 (4 channels via
// float4), row-subgroup rg = t>>6. Fully coalesced 256B-per-16-lane loads.
__global__ void ca_pool_partial(const float* __restrict__ x,
                                float* __restrict__ psum,
                                float* __restrict__ pmax) {
    const int b  = blockIdx.x;
    const int s  = blockIdx.y;
    const int t  = threadIdx.x;
    const int c4 = t & 63;
    const int rg = t >> 6;

    const f4* x4 = reinterpret_cast<const f4*>(x);
    // index in f4 units: (b*HW + hw)*64 + c4
    size_t base = ((size_t)b * HW_ + (size_t)s * ROWS_) * 64 + c4;

    f4 sum = {0.0f, 0.0f, 0.0f, 0.0f};
    f4 mx  = {-3.402823466e38f, -3.402823466e38f,
              -3.402823466e38f, -3.402823466e38f};
    #pragma unroll 7
    for (int j = 0; j < ROWS_ / 4; ++j) {           // 49 iterations
        f4 v = x4[base + (size_t)(rg + j * 4) * 64];
        sum += v;
        mx = max4(mx, v);
    }

    __shared__ f4 sred[256];
    __shared__ f4 mred[256];
    sred[t] = sum; mred[t] = mx;
    __syncthreads();
    if (rg == 0) {
        f4 s0 = sred[t] + sred[t + 64] + sred[t + 128] + sred[t + 192];
        f4 m0 = max4(max4(mred[t], mred[t + 64]),
                     max4(mred[t + 128], mred[t + 192]));
        f4* ps = reinterpret_cast<f4*>(psum);
        f4* pm = reinterpret_cast<f4*>(pmax);
        ps[(b * S_ + s) * 64 + c4] = s0;
        pm[(b * S_ + s) * 64 + c4] = m0;
    }
}

// ---- kernel 2: reduce partials into stacked pooled matrix ------------------
// P rows 0..31 = avg pool, rows 32..63 = max pool.  grid B, block C.
__global__ void ca_pool_reduce(const float* __restrict__ psum,
                               const float* __restrict__ pmax,
                               float* __restrict__ P) {
    const int b = blockIdx.x;
    const int c = threadIdx.x;
    float s = 0.0f, m = -3.402823466e38f;
    #pragma unroll
    for (int i = 0; i < S_; ++i) {
        s += psum[(b * S_ + i) * C_ + c];
        m = fmaxf(m, pmax[(b * S_ + i) * C_ + c]);
    }
    P[b * C_ + c]        = s * (1.0f / (float)HW_);
    P[(B_ + b) * C_ + c] = m;
}

// ---- kernel 3: shared MLP + sigmoid via fp32 WMMA --------------------------
// One block, 256 threads = 8 waves (wave32).
// Layer1: H1 = relu(P @ W1 + b1), P [64,256], W1 [256,32]  -> LDS
// Fold:   G = H1[0:32] + H1[32:64]          (layer 2 is linear)
// Layer2: attn = sigmoid(G @ W2 + 2*b2),   G [32,32], W2 [32,256]
__global__ void ca_mlp_attn(const float* __restrict__ P,
                            const float* __restrict__ w1,
                            const float* __restrict__ b1,
                            const float* __restrict__ w2,
                            const float* __restrict__ b2,
                            float* __restrict__ attn) {
    __shared__ float h1[64 * CR_];   // 8 KB
    __shared__ float g[32 * CR_];    // 4 KB

    const int t    = threadIdx.x;
    const int wave = t >> 5;
    const int lane = t & 31;
    const int mr   = lane & 15;        // row (A) / col (B,D) within tile
    const int kb   = (lane >> 4) * 2;  // K-pair selected by lane half
    const int dhi  = (lane >> 4) * 8;  // D-row offset for upper lane half

    // ---- layer 1: 8 tiles (mt = wave>>1 in 0..3, nt = wave&1 in 0..1) -----
    {
        const int m0 = (wave >> 1) * 16;
        const int n0 = (wave & 1) * 16;
        v8f acc = {};
        for (int k = 0; k < C_; k += 4) {
            v2f a, bm;
            a.x  = P[(m0 + mr) * C_ + k + kb];
            a.y  = P[(m0 + mr) * C_ + k + kb + 1];
            bm.x = w1[(k + kb) * CR_ + n0 + mr];
            bm.y = w1[(k + kb + 1) * CR_ + n0 + mr];
            acc = __builtin_amdgcn_wmma_f32_16x16x4_f32(
                false, a, false, bm, (short)0, acc, false, false);
        }
        const int n    = n0 + mr;
        const float bi = b1[n];
        #pragma unroll
        for (int r = 0; r < 8; ++r) {
            float h = acc[r] + bi;
            h1[(m0 + dhi + r) * CR_ + n] = fmaxf(h, 0.0f);
        }
    }
    __syncthreads();

    // ---- fold avg-half + max-half (layer 2 is linear) ---------------------
    for (int i = t; i < 32 * CR_; i += 256)
        g[i] = h1[i] + h1[i + 32 * CR_];
    __syncthreads();

    // ---- layer 2: 32 tiles (mt in 0..1, nt in 0..15), 4 per wave ----------
    for (int tt = wave * 4; tt < wave * 4 + 4; ++tt) {
        const int m0 = (tt >> 4) * 16;
        const int n0 = (tt & 15) * 16;
        v8f acc = {};
        #pragma unroll
        for (int k = 0; k < CR_; k += 4) {
            v2f a, bm;
            a.x  = g[(m0 + mr) * CR_ + k + kb];
            a.y  = g[(m0 + mr) * CR_ + k + kb + 1];
            bm.x = w2[(k + kb) * C_ + n0 + mr];
            bm.y = w2[(k + kb + 1) * C_ + n0 + mr];
            acc = __builtin_amdgcn_wmma_f32_16x16x4_f32(
                false, a, false, bm, (short)0, acc, false, false);
        }
        const int n    = n0 + mr;
        const float bi = 2.0f * b2[n];
        #pragma unroll
        for (int r = 0; r < 8; ++r) {
            float v = acc[r] + bi;
            attn[(m0 + dhi + r) * C_ + n] = 1.0f / (1.0f + expf(-v));
        }
    }
}

// ---- kernel 4: out = x * attn[b,c] ----------------------------------------
// float4 per thread; NT store so the 103 MB of writes don't evict x from the
// 192 MB L2 (x was just pulled in by the pooling pass).
__global__ void ca_scale(const float* __restrict__ x,
                         const float* __restrict__ attn,
                         float* __restrict__ out) {
    const unsigned idx = blockIdx.x * 256u + threadIdx.x;   // f4 index
    const f4* x4 = reinterpret_cast<const f4*>(x);
    const f4* a4 = reinterpret_cast<const f4*>(attn);
    f4* o4       = reinterpret_cast<f4*>(out);

    f4 xv = x4[idx];
    const unsigned b  = idx / (HW_ * 64u);   // 200704 f4 per batch
    const unsigned c4 = idx & 63u;
    f4 av = a4[b * 64u + c4];
    f4 ov = xv * av;
    __builtin_nontemporal_store(ov, o4 + idx);
}

// ---- launcher --------------------------------------------------------------
extern "C" void kernel_launch(void* const* d_in, const int* in_sizes, int n_in,
                              void* d_out, int out_size, void* d_ws, size_t ws_size,
                              hipStream_t stream) {
    const float* x  = (const float*)d_in[0];
    const float* w1 = (const float*)d_in[1];
    const float* b1 = (const float*)d_in[2];
    const float* w2 = (const float*)d_in[3];
    const float* b2 = (const float*)d_in[4];
    float* out = (float*)d_out;

    // workspace layout (floats):
    float* ws   = (float*)d_ws;
    float* psum = ws;                            // [B][S][C]   = 131072
    float* pmax = ws + (size_t)B_ * S_ * C_;     // [B][S][C]   = 131072
    float* P    = pmax + (size_t)B_ * S_ * C_;   // [64][C]     = 16384
    float* attn = P + 64 * C_;                   // [B][C]      = 8192

    ca_pool_partial<<<dim3(B_, S_), 256, 0, stream>>>(x, psum, pmax);
    ca_pool_reduce<<<B_, C_, 0, stream>>>(psum, pmax, P);
    ca_mlp_attn<<<1, 256, 0, stream>>>(P, w1, b1, w2, b2, attn);

    const int n4 = B_ * HW_ * C_ / 4;            // 6,422,528 float4
    ca_scale<<<n4 / 256, 256, 0, stream>>>(x, attn, out);
}